// Atten2Map_48344151884104
// MI455X (gfx1250) — compile-verified
//
#include <hip/hip_runtime.h>

// Problem constants from the reference
#define NB    4
#define NLOC  256
#define NNEI  128
#define NI    128
#define ND    32
#define NH    4
#define QK_COLS 256          // ND * 2 * NH

// LDS padded strides (floats) chosen for conflict-free access on 64 banks
#define G2_STRIDE 132        // 132 % 64 == 4  -> conflict-free b64 A-frag reads
#define QK_STRIDE 257        // 257 % 64 == 1  -> conflict-free strided q/k reads

#define SMEM_FLOATS (NNEI*G2_STRIDE + NNEI*QK_STRIDE + NNEI*4)
#define SMEM_BYTES  (SMEM_FLOATS * 4)

typedef float v2f __attribute__((ext_vector_type(2)));
typedef float v8f __attribute__((ext_vector_type(8)));

__global__ __launch_bounds__(256) void atten2map_kernel(
    const float* __restrict__ g2,   // (NB,NLOC,NNEI,NI)
    const float* __restrict__ h2,   // (NB,NLOC,NNEI,3)
    const float* __restrict__ Wqk,  // (NI, 256)
    float* __restrict__ out)        // (NB,NLOC,NNEI,NNEI,NH)
{
  extern __shared__ float smem[];
  float* g2s = smem;                       // [128][132]
  float* qks = g2s + NNEI * G2_STRIDE;     // [128][257]
  float* h2s = qks + NNEI * QK_STRIDE;     // [128][4]

  const int blk  = blockIdx.x;             // b*NLOC + l
  const int tid  = threadIdx.x;
  const int wave = tid >> 5;               // 0..7
  const int lane = tid & 31;
  const int lh   = lane >> 4;              // lane half (K-half / row-half select)
  const int ll   = lane & 15;              // lane within half (M / N index)

  // ---------------- Phase A: stage g2 tile + h2 into LDS ----------------
  {
    const float* g2g = g2 + (size_t)blk * (NNEI * NI);
    #pragma unroll
    for (int i = 0; i < 16; ++i) {
      int e = i * 1024 + tid * 4;          // 16384 floats total
      int r = e >> 7, c = e & 127;
      float4 v = *(const float4*)(g2g + e);
      *(float4*)(g2s + r * G2_STRIDE + c) = v;
    }
    if (tid < NNEI) {
      const float* hp = h2 + (size_t)blk * (NNEI * 3) + tid * 3;
      h2s[tid * 4 + 0] = hp[0];
      h2s[tid * 4 + 1] = hp[1];
      h2s[tid * 4 + 2] = hp[2];
      h2s[tid * 4 + 3] = 0.0f;
    }
  }
  __syncthreads();

  // ---------------- Phase B: qk = g2 @ Wqk  (128x128 @ 128x256) ----------
  // Wave w owns output column-tiles {2w, 2w+1}; 8 row-tiles each.
  for (int cti = 0; cti < 2; ++cti) {
    const int n0 = (wave * 2 + cti) * 16;
    v8f acc[8] = {};
    for (int ks = 0; ks < 32; ++ks) {
      const int k0 = ks * 4 + 2 * lh;      // this lane's K pair
      v2f bfrag;                           // B: 4x16 of Wqk
      bfrag.x = Wqk[(k0 + 0) * QK_COLS + n0 + ll];
      bfrag.y = Wqk[(k0 + 1) * QK_COLS + n0 + ll];
      #pragma unroll
      for (int rt = 0; rt < 8; ++rt) {     // A: 16x4 of g2 from LDS
        v2f afrag = *(const v2f*)(g2s + (rt * 16 + ll) * G2_STRIDE + k0);
        acc[rt] = __builtin_amdgcn_wmma_f32_16x16x4_f32(
            false, afrag, false, bfrag, (short)0, acc[rt], false, false);
      }
    }
    #pragma unroll
    for (int rt = 0; rt < 8; ++rt) {
      #pragma unroll
      for (int v = 0; v < 8; ++v) {        // C layout: row = v + 8*lh, col = ll
        qks[(rt * 16 + v + 8 * lh) * QK_STRIDE + n0 + ll] = acc[rt][v];
      }
    }
  }
  __syncthreads();

  // ---------------- Phase C: scores, softmax, * h2h2t, store --------------
  const float qscale = 0.17677669529663687f;  // 1/sqrt(ND)
  const float hscale = 0.57735026918962576f;  // 1/sqrt(3)
  float* outp = out + (size_t)blk * (NNEI * NNEI * NH);

  // per-lane h2 for columns m = t*16 + ll (same for all strips)
  float hmx[8], hmy[8], hmz[8];
  #pragma unroll
  for (int t = 0; t < 8; ++t) {
    const float* hp = h2s + (t * 16 + ll) * 4;
    hmx[t] = hp[0]; hmy[t] = hp[1]; hmz[t] = hp[2];
  }

  for (int s = 0; s < 4; ++s) {
    const int sid   = wave * 4 + s;        // 0..31 = 4 heads x 8 n-tiles
    const int h     = sid >> 3;
    const int nrow0 = (sid & 7) * 16;

    // scores strip: rows nrow0..+15, all 128 cols, head h
    v8f acc[8] = {};
    for (int ks = 0; ks < 8; ++ks) {       // K loop over d (ND=32, 4/step)
      const int d0 = ks * 4 + 2 * lh;
      v2f afrag;                           // q[n][d] = qk[n][d*8 + h]
      {
        const float* qp = qks + (nrow0 + ll) * QK_STRIDE + d0 * 8 + h;
        afrag.x = qp[0];
        afrag.y = qp[8];
      }
      #pragma unroll
      for (int t = 0; t < 8; ++t) {        // B[d][m] = k[m][d] = qk[m][d*8+4+h]
        v2f bfrag;
        const float* kp = qks + (t * 16 + ll) * QK_STRIDE + d0 * 8 + 4 + h;
        bfrag.x = kp[0];
        bfrag.y = kp[8];
        acc[t] = __builtin_amdgcn_wmma_f32_16x16x4_f32(
            false, afrag, false, bfrag, (short)0, acc[t], false, false);
      }
    }

    // scale
    #pragma unroll
    for (int t = 0; t < 8; ++t) acc[t] = acc[t] * qscale;

    // softmax over m: row r = v + 8*lh lives in one lane-half (16 lanes x 8 tiles)
    float rinv[8];
    #pragma unroll
    for (int v = 0; v < 8; ++v) {
      float m = acc[0][v];
      #pragma unroll
      for (int t = 1; t < 8; ++t) m = fmaxf(m, acc[t][v]);
      #pragma unroll
      for (int off = 1; off < 16; off <<= 1)
        m = fmaxf(m, __shfl_xor(m, off, 32));
      float ssum = 0.0f;
      #pragma unroll
      for (int t = 0; t < 8; ++t) {
        float e = __expf(acc[t][v] - m);
        acc[t][v] = e;
        ssum += e;
      }
      #pragma unroll
      for (int off = 1; off < 16; off <<= 1)
        ssum += __shfl_xor(ssum, off, 32);
      rinv[v] = 1.0f / ssum;
    }

    // per-lane h2 for rows n = nrow0 + v + 8*lh
    float hnx[8], hny[8], hnz[8];
    #pragma unroll
    for (int v = 0; v < 8; ++v) {
      const float* hp = h2s + (nrow0 + v + 8 * lh) * 4;
      hnx[v] = hp[0]; hny[v] = hp[1]; hnz[v] = hp[2];
    }

    // out[n][m][h] = softmax * (h2[n].h2[m])/sqrt(3)
    #pragma unroll
    for (int t = 0; t < 8; ++t) {
      const int m = t * 16 + ll;
      #pragma unroll
      for (int v = 0; v < 8; ++v) {
        const int n = nrow0 + v + 8 * lh;
        float hh = (hnx[v] * hmx[t] + hny[v] * hmy[t] + hnz[v] * hmz[t]) * hscale;
        outp[((size_t)n * NNEI + m) * NH + h] = acc[t][v] * rinv[v] * hh;
      }
    }
  }
}

extern "C" void kernel_launch(void* const* d_in, const int* in_sizes, int n_in,
                              void* d_out, int out_size, void* d_ws, size_t ws_size,
                              hipStream_t stream) {
  const float* g2  = (const float*)d_in[0];
  const float* h2  = (const float*)d_in[1];
  const float* Wqk = (const float*)d_in[2];
  float* out = (float*)d_out;
  (void)in_sizes; (void)n_in; (void)out_size; (void)d_ws; (void)ws_size;

  dim3 grid(NB * NLOC);   // 1024 workgroups, one per (b, l)
  dim3 block(256);        // 8 wave32s
  atten2map_kernel<<<grid, block, SMEM_BYTES, stream>>>(g2, h2, Wqk, out);
}